// GCN_6167573037327
// MI455X (gfx1250) — compile-verified
//
#include <hip/hip_runtime.h>
#include <hip/hip_bf16.h>

// ---------------------------------------------------------------------------
// 3-layer GCN for MI455X (gfx1250, wave32).
//  - Dense per-node linear transform via V_WMMA_F32_16X16X4_F32 (exact f32).
//  - Edge scatter-add via hardware global_atomic_add_f32 (unsafeAtomicAdd).
//  - A-tiles staged in LDS with padded stride (132 floats) for conflict-free
//    ds_load_b64 fragment reads.
// Scatter/atomic traffic dominates (≈5 GB L2 traffic, ~600M atomics) so the
// GEMM precision choice is free; keep f32 to match the reference exactly.
// ---------------------------------------------------------------------------

typedef float v2f __attribute__((ext_vector_type(2)));
typedef float v8f __attribute__((ext_vector_type(8)));

#define F 128           // feature width (both in and out of every layer)
#define AS_STRIDE 132   // padded LDS row stride (floats); even & mult-of-4

// ---------------- degree / normalization ----------------------------------

__global__ __launch_bounds__(256) void k_init_deg(float* deg, int n) {
    int i = blockIdx.x * blockDim.x + threadIdx.x;
    if (i < n) deg[i] = 1.0f;               // self-loop contributes 1
}

__global__ __launch_bounds__(256) void k_accum_deg(const int* __restrict__ rows,
                                                   float* deg, int e) {
    int i = blockIdx.x * blockDim.x + threadIdx.x;
    if (i < e) unsafeAtomicAdd(&deg[rows[i]], 1.0f);
}

__global__ __launch_bounds__(256) void k_finalize_isd(float* deg_isd, int n) {
    int i = blockIdx.x * blockDim.x + threadIdx.x;
    if (i < n) deg_isd[i] = rsqrtf(deg_isd[i]);   // deg >= 1 always
}

// ---------------- weight transpose (Wt[n][k] = W[k][n]) --------------------

__global__ __launch_bounds__(256) void k_transpose_w(const float* __restrict__ W,
                                                     float* __restrict__ Wt) {
    int i = blockIdx.x * blockDim.x + threadIdx.x;   // 0 .. 16383
    if (i < F * F) {
        int k = i >> 7;          // row of W (fan_in)
        int n = i & (F - 1);     // col of W (fan_out)
        Wt[n * F + k] = W[i];
    }
}

// ---------------- WMMA GEMM: h = (relu?)(in) @ W ---------------------------
// Block: 256 threads = 8 waves. Each block computes a 16-row tile of h;
// wave w computes output columns [16w, 16w+16). K = 128 -> 32 WMMA ops.

__global__ __launch_bounds__(256) void k_gemm_wmma(const float* __restrict__ in,
                                                   const float* __restrict__ Wt,
                                                   float* __restrict__ h,
                                                   int n, int relu_in) {
    __shared__ float As[16 * AS_STRIDE];   // 8448 B

    const int tid  = threadIdx.x;
    const int wave = tid >> 5;
    const int lane = tid & 31;
    const int block_row = blockIdx.x * 16;

    // Cooperative staging of the 16x128 A tile (coalesced float4 reads).
    for (int i = tid; i < 16 * 32; i += 256) {
        int r  = i >> 5;
        int c4 = (i & 31) << 2;
        int row = block_row + r;
        float4 v = make_float4(0.f, 0.f, 0.f, 0.f);
        if (row < n) v = *(const float4*)(in + (size_t)row * F + c4);
        if (relu_in) {
            v.x = fmaxf(v.x, 0.f); v.y = fmaxf(v.y, 0.f);
            v.z = fmaxf(v.z, 0.f); v.w = fmaxf(v.w, 0.f);
        }
        float* dst = &As[r * AS_STRIDE + c4];
        dst[0] = v.x; dst[1] = v.y; dst[2] = v.z; dst[3] = v.w;
    }
    __syncthreads();

    const int m     = lane & 15;     // row (A) / col (B) within 16
    const int hi    = lane >> 4;     // half-wave selects K pair
    const int nbase = wave * 16;     // output column tile of this wave

    v8f acc = {};
    for (int k = 0; k < F; k += 4) {
        const int kk = k + (hi << 1);          // K, K+1 slice for this lane
        v2f a = *(const v2f*)(&As[m * AS_STRIDE + kk]);                 // A[m][kk..kk+1]
        v2f b = *(const v2f*)(Wt + (size_t)(nbase + m) * F + kk);       // B[kk..kk+1][nbase+m]
        acc = __builtin_amdgcn_wmma_f32_16x16x4_f32(
            /*neg_a=*/false, a, /*neg_b=*/false, b,
            /*c_mod=*/(short)0, acc, /*reuse_a=*/false, /*reuse_b=*/false);
    }

    // C/D layout: lane<16 -> rows 0..7, lane>=16 -> rows 8..15; N = nbase+m.
    const int row0 = block_row + hi * 8;
    const int ncol = nbase + m;
    #pragma unroll
    for (int j = 0; j < 8; ++j) {
        int row = row0 + j;
        if (row < n) h[(size_t)row * F + ncol] = acc[j];
    }
}

// ---------------- out = bias + isd[n]^2 * h[n]  (self-loop fused) ----------

__global__ __launch_bounds__(256) void k_init_out(const float* __restrict__ h,
                                                  const float* __restrict__ isd,
                                                  const float* __restrict__ bias,
                                                  float* __restrict__ out, int n) {
    int idx = blockIdx.x * blockDim.x + threadIdx.x;   // n*32 threads
    if (idx >= n * 32) return;
    int node = idx >> 5;
    int c4   = (idx & 31) << 2;
    float s = isd[node]; s *= s;
    float4 hv = *(const float4*)(h + (size_t)node * F + c4);
    float4 o;
    o.x = bias[c4 + 0] + s * hv.x;
    o.y = bias[c4 + 1] + s * hv.y;
    o.z = bias[c4 + 2] + s * hv.z;
    o.w = bias[c4 + 3] + s * hv.w;
    *(float4*)(out + (size_t)node * F + c4) = o;
}

// ---------------- edge scatter: out[row] += norm * h[col] ------------------
// One wave (32 lanes) per edge; each lane handles 4 contiguous floats.

__global__ __launch_bounds__(256) void k_agg_edges(const int* __restrict__ rows,
                                                   const int* __restrict__ cols,
                                                   const float* __restrict__ isd,
                                                   const float* __restrict__ h,
                                                   float* __restrict__ out, int e) {
    int gwave = (blockIdx.x * blockDim.x + threadIdx.x) >> 5;
    int lane  = threadIdx.x & 31;
    if (gwave >= e) return;
    int r = rows[gwave];
    int c = cols[gwave];
    float nrm = isd[r] * isd[c];
    float4 v = *((const float4*)(h + (size_t)c * F) + lane);   // global_load_b128
    float* o = out + (size_t)r * F + (lane << 2);
    unsafeAtomicAdd(o + 0, v.x * nrm);
    unsafeAtomicAdd(o + 1, v.y * nrm);
    unsafeAtomicAdd(o + 2, v.z * nrm);
    unsafeAtomicAdd(o + 3, v.w * nrm);
}

// ---------------------------------------------------------------------------

extern "C" void kernel_launch(void* const* d_in, const int* in_sizes, int n_in,
                              void* d_out, int out_size, void* d_ws, size_t ws_size,
                              hipStream_t stream) {
    const float* x  = (const float*)d_in[0];
    const int*   ei = (const int*)d_in[1];          // [2, E] flat: rows then cols
    const float* W[3] = { (const float*)d_in[2], (const float*)d_in[4], (const float*)d_in[6] };
    const float* B[3] = { (const float*)d_in[3], (const float*)d_in[5], (const float*)d_in[7] };
    float* out = (float*)d_out;

    const int N = in_sizes[0] / F;
    const int E = in_sizes[1] / 2;
    const int* rows = ei;
    const int* cols = ei + E;

    // workspace layout: isd | h | Wt
    float* isd  = (float*)d_ws;
    float* hbuf = isd + ((N + 63) & ~63);           // 256B-aligned
    float* Wt   = hbuf + (size_t)N * F;

    // --- degree / normalization ---
    k_init_deg   <<<(N + 255) / 256, 256, 0, stream>>>(isd, N);
    k_accum_deg  <<<(E + 255) / 256, 256, 0, stream>>>(rows, isd, E);
    k_finalize_isd<<<(N + 255) / 256, 256, 0, stream>>>(isd, N);

    const int gemm_blocks = (N + 15) / 16;
    const int init_blocks = (N * 32 + 255) / 256;
    const int agg_blocks  = (E + 7) / 8;            // 8 waves/block, 1 edge/wave

    // --- layer 1: in = x, out -> d_out ---
    k_transpose_w<<<(F * F + 255) / 256, 256, 0, stream>>>(W[0], Wt);
    k_gemm_wmma  <<<gemm_blocks, 256, 0, stream>>>(x, Wt, hbuf, N, 0);
    k_init_out   <<<init_blocks, 256, 0, stream>>>(hbuf, isd, B[0], out, N);
    k_agg_edges  <<<agg_blocks, 256, 0, stream>>>(rows, cols, isd, hbuf, out, E);

    // --- layer 2: in = relu(d_out), out -> d_out ---
    k_transpose_w<<<(F * F + 255) / 256, 256, 0, stream>>>(W[1], Wt);
    k_gemm_wmma  <<<gemm_blocks, 256, 0, stream>>>(out, Wt, hbuf, N, 1);
    k_init_out   <<<init_blocks, 256, 0, stream>>>(hbuf, isd, B[1], out, N);
    k_agg_edges  <<<agg_blocks, 256, 0, stream>>>(rows, cols, isd, hbuf, out, E);

    // --- layer 3: in = relu(d_out), out -> d_out (no final relu) ---
    k_transpose_w<<<(F * F + 255) / 256, 256, 0, stream>>>(W[2], Wt);
    k_gemm_wmma  <<<gemm_blocks, 256, 0, stream>>>(out, Wt, hbuf, N, 1);
    k_init_out   <<<init_blocks, 256, 0, stream>>>(hbuf, isd, B[2], out, N);
    k_agg_edges  <<<agg_blocks, 256, 0, stream>>>(rows, cols, isd, hbuf, out, E);
}